// ClusteringLoss_17145509446225
// MI455X (gfx1250) — compile-verified
//
#include <hip/hip_runtime.h>
#include <math.h>

typedef __attribute__((ext_vector_type(2))) float v2f;
typedef __attribute__((ext_vector_type(8))) float v8f;

#define BB 32
#define PP (360 * 640)      // 230400 pixels per batch
#define NCHUNK 96
#define PIXB (PP / NCHUNK)  // 2400 pixels per block
#define D_V 1.0f
#define D_D 6.0f

// ws layout (float index):
//   [0,256)      counts  [B][8]        (float-exact integer counts)
//   [256,1280)   sums    [B][8][4]
//   [1280,2304)  meansx  [B][8][4]     (means_ext; slots 0,6,7 == 0)
//   [2304,2560)  validf  [B][8]        (1.0 iff lane slot 1..5 and count>1)
//   [2560,2562)  accum   {dist_sum, point_count}

__global__ void k_zero(float* ws) {
    for (int i = threadIdx.x; i < 2562; i += 256) ws[i] = 0.f;
}

// Pass 1: segmented sums/counts. Per-thread register accumulators (exact f32),
// wave32 butterfly reduce, LDS combine across 8 waves, global f32 atomics.
__global__ void k_accum(const int* __restrict__ tgt, const float* __restrict__ emb,
                        float* __restrict__ counts, float* __restrict__ sums) {
    __shared__ float sbin[25];
    const int tid = threadIdx.x;
    const int b   = blockIdx.y;
    if (tid < 25) sbin[tid] = 0.f;
    __syncthreads();

    float v[25];
#pragma unroll
    for (int i = 0; i < 25; ++i) v[i] = 0.f;

    const int pbeg = blockIdx.x * PIXB;
    const int pend = pbeg + PIXB;
    const int tb   = b * PP;
    const int eb   = b * 4 * PP;
    for (int p = pbeg + tid; p < pend; p += 256) {
        const int t  = tgt[tb + p];
        const float e0 = emb[eb + p];
        const float e1 = emb[eb + PP + p];
        const float e2 = emb[eb + 2 * PP + p];
        const float e3 = emb[eb + 3 * PP + p];
#pragma unroll
        for (int l = 1; l <= 5; ++l) {
            const float m = (t == l) ? 1.f : 0.f;   // cmp + cndmask, then 5 FMAs
            v[(l - 1) * 4 + 0] += m * e0;
            v[(l - 1) * 4 + 1] += m * e1;
            v[(l - 1) * 4 + 2] += m * e2;
            v[(l - 1) * 4 + 3] += m * e3;
            v[20 + (l - 1)]    += m;
        }
    }
    // wave32 butterfly reduction
#pragma unroll
    for (int off = 16; off > 0; off >>= 1) {
#pragma unroll
        for (int i = 0; i < 25; ++i) v[i] += __shfl_xor(v[i], off, 32);
    }
    if ((tid & 31) == 0) {
#pragma unroll
        for (int i = 0; i < 25; ++i) atomicAdd(&sbin[i], v[i]);
    }
    __syncthreads();
    if (tid < 20)      atomicAdd(&sums[b * 32 + 4 + tid], sbin[tid]);        // slots 1..5, ch 0..3
    else if (tid < 25) atomicAdd(&counts[b * 8 + 1 + (tid - 20)], sbin[tid]);
}

// Means / validity table + point_count. One block of 256 = B*8 threads.
__global__ void k_means(const float* __restrict__ counts, const float* __restrict__ sums,
                        float* __restrict__ meansx, float* __restrict__ validf,
                        float* __restrict__ accum) {
    __shared__ float spc;
    const int tid = threadIdx.x;
    if (tid == 0) spc = 0.f;
    __syncthreads();
    const int b = tid >> 3, s = tid & 7;
    const float cnt = counts[b * 8 + s];
    const bool lane_slot = (s >= 1 && s <= 5);
    const bool valid = lane_slot && (cnt > 1.5f);   // counts are exact integers
    const float inv = 1.f / fmaxf(cnt, 1.f);
#pragma unroll
    for (int c = 0; c < 4; ++c)
        meansx[b * 32 + s * 4 + c] = lane_slot ? sums[b * 32 + s * 4 + c] * inv : 0.f;
    validf[b * 8 + s] = valid ? 1.f : 0.f;
    float pc = valid ? cnt : 0.f;
#pragma unroll
    for (int off = 16; off > 0; off >>= 1) pc += __shfl_xor(pc, off, 32);
    if ((tid & 31) == 0) atomicAdd(&spc, pc);
    __syncthreads();
    if (tid == 0) accum[1] = spc;
}

// Pass 2: per-pixel hinge distance to own lane mean (means broadcast from LDS).
__global__ void k_dist(const int* __restrict__ tgt, const float* __restrict__ emb,
                       const float* __restrict__ meansx, const float* __restrict__ validf,
                       float* __restrict__ accum) {
    __shared__ float lm[32];
    __shared__ float lv[8];
    __shared__ float sacc;
    const int tid = threadIdx.x;
    const int b   = blockIdx.y;
    if (tid < 32) lm[tid] = meansx[b * 32 + tid];
    if (tid < 8)  lv[tid] = validf[b * 8 + tid];
    if (tid == 0) sacc = 0.f;
    __syncthreads();

    const int pbeg = blockIdx.x * PIXB;
    const int pend = pbeg + PIXB;
    const int tb   = b * PP;
    const int eb   = b * 4 * PP;
    float local = 0.f;
    for (int p = pbeg + tid; p < pend; p += 256) {
        const int t  = tgt[tb + p];
        const float e0 = emb[eb + p];
        const float e1 = emb[eb + PP + p];
        const float e2 = emb[eb + 2 * PP + p];
        const float e3 = emb[eb + 3 * PP + p];
        const float d0 = e0 - lm[t * 4 + 0];
        const float d1 = e1 - lm[t * 4 + 1];
        const float d2 = e2 - lm[t * 4 + 2];
        const float d3 = e3 - lm[t * 4 + 3];
        const float sq = d0 * d0 + d1 * d1 + d2 * d2 + d3 * d3;
        const float dist = sqrtf(fmaxf(sq, 1e-12f));
        const float h = fmaxf(dist - D_V, 0.f);
        local += lv[t] * h * h;
    }
#pragma unroll
    for (int off = 16; off > 0; off >>= 1) local += __shfl_xor(local, off, 32);
    if ((tid & 31) == 0) atomicAdd(&sacc, local);
    __syncthreads();
    if (tid == 0) atomicAdd(&accum[0], sacc);
}

// Final: per-batch Gram G = M*M^T via V_WMMA_F32_16X16X4_F32 (M=[16x4] means,
// rows 0..4 = lanes 1..5, K=C=4 — exact f32). psq(i,j)=G[ii]+G[jj]-2G[ij].
// Single wave32, no divergence before/around the WMMA (EXEC must be all 1s).
__global__ void __launch_bounds__(32) k_final(const float* __restrict__ meansx,
                                              const float* __restrict__ validf,
                                              const float* __restrict__ accum,
                                              float* __restrict__ out) {
    const int lane = threadIdx.x;        // 0..31
    const int row  = lane & 15;          // A: M index / B: N index
    const int g2   = lane >> 4;          // K half: holds K = 2*g2, 2*g2+1
    const int s    = (row < 5) ? (row + 1) : 0;   // slot 0 is all-zero padding
    float var_sum = 0.f, nb = 0.f;

    for (int b = 0; b < BB; ++b) {
        // A(16x4) lane layout: lane m(+16) holds K=2*g2, 2*g2+1 of row m.
        // B(4x16) lane layout mirrors it; for the symmetric Gram the per-lane
        // register contents of A and B coincide, so pass the same operand.
        v2f a;
        a.x = meansx[b * 32 + s * 4 + 2 * g2 + 0];
        a.y = meansx[b * 32 + s * 4 + 2 * g2 + 1];
        v8f cz = {0.f, 0.f, 0.f, 0.f, 0.f, 0.f, 0.f, 0.f};
        v8f gm = __builtin_amdgcn_wmma_f32_16x16x4_f32(
            /*neg_a=*/false, a, /*neg_b=*/false, a,
            /*c_mod=*/(short)0, cz, /*reuse_a=*/false, /*reuse_b=*/false);
        // D layout: lane l, vgpr r -> G[r + 8*(l>>4)][l&15]
        const float diag0 = __shfl(gm[0], 0, 32);
        const float diag1 = __shfl(gm[1], 1, 32);
        const float diag2 = __shfl(gm[2], 2, 32);
        const float diag3 = __shfl(gm[3], 3, 32);
        const float diag4 = __shfl(gm[4], 4, 32);
        const float diag[5] = {diag0, diag1, diag2, diag3, diag4};
        float mydiag = diag0;
        mydiag = (lane == 1) ? diag1 : mydiag;
        mydiag = (lane == 2) ? diag2 : mydiag;
        mydiag = (lane == 3) ? diag3 : mydiag;
        mydiag = (lane == 4) ? diag4 : mydiag;

        const float myv = validf[b * 8 + s] * ((lane < 5) ? 1.f : 0.f);
        float vr[5];
#pragma unroll
        for (int r = 0; r < 5; ++r) vr[r] = __shfl(myv, r, 32);

        float colsum = 0.f, colcnt = 0.f;
#pragma unroll
        for (int r = 0; r < 5; ++r) {       // pair (i=r, j=lane), r < lane < 5
            const float mask = ((lane < 5) && (r < lane)) ? (vr[r] * myv) : 0.f;
            const float psq  = diag[r] + mydiag - 2.f * gm[r];
            const float pd   = sqrtf(fmaxf(psq, 1e-12f));
            const float ph   = fmaxf(D_D - pd, 0.f);
            colsum += mask * ph * ph;
            colcnt += mask;
        }
#pragma unroll
        for (int off = 16; off > 0; off >>= 1) {
            colsum += __shfl_xor(colsum, off, 32);
            colcnt += __shfl_xor(colcnt, off, 32);
        }
        const float has = (colcnt > 0.f) ? 1.f : 0.f;
        var_sum += has * colsum / fmaxf(colcnt, 1.f);
        nb      += has;
    }

    if (lane == 0) {
        const float dsum = accum[0], pc = accum[1];
        const float dist_loss = (pc > 0.f) ? dsum / fmaxf(pc, 1.f) : 0.f;
        const float var_loss  = (nb > 0.f) ? var_sum / fmaxf(nb, 1.f) : 0.f;
        out[0] = dist_loss + var_loss;
    }
}

extern "C" void kernel_launch(void* const* d_in, const int* in_sizes, int n_in,
                              void* d_out, int out_size, void* d_ws, size_t ws_size,
                              hipStream_t stream) {
    (void)in_sizes; (void)n_in; (void)out_size; (void)ws_size;
    const int*   targets = (const int*)d_in[0];
    const float* emb     = (const float*)d_in[1];
    float* ws     = (float*)d_ws;
    float* counts = ws;
    float* sums   = ws + 256;
    float* meansx = ws + 1280;
    float* validf = ws + 2304;
    float* accum  = ws + 2560;
    float* out    = (float*)d_out;

    k_zero<<<1, 256, 0, stream>>>(ws);
    k_accum<<<dim3(NCHUNK, BB), 256, 0, stream>>>(targets, emb, counts, sums);
    k_means<<<1, 256, 0, stream>>>(counts, sums, meansx, validf, accum);
    k_dist<<<dim3(NCHUNK, BB), 256, 0, stream>>>(targets, emb, meansx, validf, accum);
    k_final<<<1, 32, 0, stream>>>(meansx, validf, accum, out);
}